// DMV_27315992002738
// MI455X (gfx1250) — compile-verified
//
#include <hip/hip_runtime.h>
#include <hip/hip_bf16.h>

typedef __attribute__((ext_vector_type(2))) float v2f;
typedef __attribute__((ext_vector_type(8))) float v8f;

#define NEGV  (-1e9f)
#define BIGP  (3.0e9f)       // shift penalty: exp(x - (s+BIGP)) == 0 for any chart value
#define L2E   (1.4426950408889634f)
#define NN    41
#define STR   42
#define CVV   2

#if defined(__has_builtin)
#  if __has_builtin(__builtin_amdgcn_wmma_f32_16x16x4_f32)
#    define HAVE_WMMA4 1
#  endif
#  if __has_builtin(__builtin_amdgcn_exp2f)
#    define EXP2I(x) __builtin_amdgcn_exp2f(x)
#  endif
#endif
#ifndef HAVE_WMMA4
#  define HAVE_WMMA4 0
#endif
#ifndef EXP2I
#  define EXP2I(x) __expf((x) * 0.6931471805599453f)
#endif

// exp(x - shift), with shiftC = shift * log2(e): one v_fma + one v_exp.
__device__ __forceinline__ float exp_shift(float x, float shiftC) {
  return EXP2I(fmaf(x, L2E, -shiftC));
}

__device__ __forceinline__ float logadd(float a, float b) {
  float m = fmaxf(a, b);
  float n = fminf(a, b);
  return m + __logf(1.0f + __expf(n - m));   // n - m <= 0, NaN-free
}

// For a tile of up to 16 spans starting at i0, compute
//   D[d] = logsumexp_{t=0..L-1}  U[i0+d][i0+d+uoff+t] + V[i0+d+voff+t][i0+d+w]
// via exp-domain banded matmul on the WMMA unit (diagonal of a 16x16xL product).
// Row validity and K-tail masking are folded ADDITIVELY into the exp argument
// so every LDS load is unconditional (no exec-masked regions in the hot loop).
__device__ __forceinline__ float band_lse_tile(
    const float* U, const float* V,
    int i0, int s, int w, int L, int uoff, int voff,
    int lane, bool& hasD)
{
  const int d    = lane & 15;
  const int row  = i0 + d;
  const bool rowOK = row < s;
  const int rowc = rowOK ? row : (s - 1);         // clamped -> safe addresses
  const int Lm1  = (L > 0) ? (L - 1) : 0;

  const float* ubase = U + rowc * STR + (rowc + uoff);          // step 1
  const float* vbase = V + (rowc + voff) * STR + (rowc + w);    // step STR

  // Max-shift pass: lanes 0-15 row-max of U (shiftA[d]); lanes 16-31 col-max of V (shiftB[d]).
  const bool isA = lane < 16;
  const float* mp    = isA ? ubase : vbase;
  const int    mstep = isA ? 1 : STR;
  float mx = NEGV;
  for (int t = 0; t < L; ++t) { mx = fmaxf(mx, *mp); mp += mstep; }
  mx = rowOK ? mx : NEGV;
  const float sa = __shfl(mx, d, 32);        // shiftA[d]
  const float sb = __shfl(mx, d + 16, 32);   // shiftB[d]
  // Fold row validity into the (pre-scaled) shifts: invalid rows -> exp == 0.
  const float saC = (rowOK ? sa : (sa + BIGP)) * L2E;
  const float sbC = (rowOK ? sb : (sb + BIGP)) * L2E;

  float diag;
#if HAVE_WMMA4
  v8f acc0 = {};
  v8f acc1 = {};
  const int khalf = (lane >> 4) << 1;        // f32 A/B layout: K = 2*(lane>=16) + vgpr
  int t0 = 0;
  // steady state: both K-chunks fully in range -> zero masking, direct addressing
  for (; t0 + 8 <= L; t0 += 8) {
    v2f a0, b0, a1, b1;
#pragma unroll
    for (int v = 0; v < 2; ++v) {
      const int k0 = t0 + khalf + v;
      const int k1 = k0 + 4;
      a0[v] = exp_shift(ubase[k0], saC);         // A[M=lane&15][k0]
      b0[v] = exp_shift(vbase[k0 * STR], sbC);   // B[k0][N=lane&15]
      a1[v] = exp_shift(ubase[k1], saC);
      b1[v] = exp_shift(vbase[k1 * STR], sbC);
    }
    acc0 = __builtin_amdgcn_wmma_f32_16x16x4_f32(
        false, a0, false, b0, (short)0, acc0, false, false);
    acc1 = __builtin_amdgcn_wmma_f32_16x16x4_f32(
        false, a1, false, b1, (short)0, acc1, false, false);
  }
  // tail: at most 2 chunks; additive value-side masking keeps loads unconditional
  for (; t0 < L; t0 += 4) {
    v2f av, bv;
#pragma unroll
    for (int v = 0; v < 2; ++v) {
      const int  kk  = t0 + khalf + v;
      const int  kkc = min(kk, Lm1);
      const float pen = (kk < L) ? 0.0f : -BIGP;
      av[v] = exp_shift(ubase[kkc] + pen, saC);
      bv[v] = exp_shift(vbase[kkc * STR] + pen, sbC);
    }
    acc0 = __builtin_amdgcn_wmma_f32_16x16x4_f32(
        false, av, false, bv, (short)0, acc0, false, false);
  }
  // C/D layout: VGPR r, lanes 0-15 -> (M=r,N=lane); lanes 16-31 -> (M=r+8,N=lane-16).
  // Diagonal lives in lanes 0-7 (VGPR=lane) and lanes 24-31 (VGPR=lane-24).
  hasD = (lane < 8) || (lane >= 24);
  const int r = (lane < 8) ? lane : (lane - 24);
  v8f accs;
#pragma unroll
  for (int q = 0; q < 8; ++q) accs[q] = acc0[q] + acc1[q];
  diag = accs[0];
#pragma unroll
  for (int q = 1; q < 8; ++q) diag = (r == q) ? accs[q] : diag;
#else
  hasD = (lane < 16);
  float ssum = 0.0f;
  for (int t = 0; t < L; ++t)
    ssum += exp_shift(ubase[t], saC) * exp_shift(vbase[t * STR], sbC);
  ssum = (hasD && rowOK) ? ssum : 0.0f;
  diag = ssum;
#endif
  return (diag > 0.0f) ? (__logf(diag) + sa + sb) : (2.0f * NEGV);
}

__launch_bounds__(256)
__global__ void dmv_inside_kernel(const float* __restrict__ trans,
                                  const float* __restrict__ dec,
                                  const int* __restrict__ lens,
                                  float* __restrict__ out)
{
  __shared__ __align__(16) float IR[NN * STR];
  __shared__ __align__(16) float IL[NN * STR];
  __shared__ __align__(16) float KR[NN * STR];
  __shared__ __align__(16) float KL[NN * STR];
  __shared__ __align__(16) float CRt[NN * STR];
  __shared__ __align__(16) float CLt[NN * STR];
  __shared__ __align__(16) float TR[NN * NN * CVV];   // trans tile for this sentence
  __shared__ __align__(16) float DC[NN * 8];          // dec tile: [pos][dir][dv][go/stop]

  const int b    = blockIdx.x;
  const int tid  = threadIdx.x;
  const int wave = tid >> 5;
  const int lane = tid & 31;

  // ---- load sentence data (vectorized) + init chart ----
  {
    // trans: 3362 floats, per-sentence base 8B-aligned -> float2
    const float2* tb2 = (const float2*)(trans + (size_t)b * (NN * NN * CVV));
    float2* tr2 = (float2*)TR;
    for (int p = tid; p < (NN * NN * CVV) / 2; p += 256) tr2[p] = tb2[p];
    // dec: 328 floats, base 16B-aligned -> float4
    const float4* db4 = (const float4*)(dec + (size_t)b * (NN * 8));
    float4* dc4 = (float4*)DC;
    for (int p = tid; p < (NN * 8) / 4; p += 256) dc4[p] = db4[p];
    // chart init: 861 float2 per table
    const float2 neg2 = make_float2(NEGV, NEGV);
    for (int p = tid; p < (NN * STR) / 2; p += 256) {
      ((float2*)IR)[p]  = neg2; ((float2*)IL)[p]  = neg2;
      ((float2*)KR)[p]  = neg2; ((float2*)KL)[p]  = neg2;
      ((float2*)CRt)[p] = neg2; ((float2*)CLt)[p] = neg2;
    }
  }
  __syncthreads();
  if (tid < NN) {
    KR[tid * STR + tid]  = 0.0f;
    KL[tid * STR + tid]  = 0.0f;
    CRt[tid * STR + tid] = DC[tid * 8 + 5];  // stop[pos][RIGHT][NOCHILD]
    CLt[tid * STR + tid] = DC[tid * 8 + 1];  // stop[pos][LEFT][NOCHILD]
  }
  __syncthreads();

  // ---- chart loop over span width ----
  for (int w = 1; w < NN; ++w) {
    const int s = NN - w;
    const int ntiles = (s + 15) >> 4;
    const int njobs = 2 * ntiles;

    // stage 1: incomplete spans IR, IL (consume widths < w only)
    for (int job = wave; job < njobs; job += 8) {
      const int table = (job >= ntiles) ? 1 : 0;
      const int i0 = (table ? (job - ntiles) : job) << 4;
      bool hasD;
      const int d = lane & 15;
      if (table == 0) {
        // IR: lse_k KR[i][k] + CL[k+1][j], k = i+1..j-1  (valence 1 part)
        float Dv = band_lse_tile(KR, CLt, i0, s, w, w - 1, 1, 2, lane, hasD);
        const int i = i0 + d;
        if (hasD && i < s) {
          const int j = i + w;
          const float base  = Dv + DC[i * 8 + 6] + TR[(i * NN + j) * 2 + 1];   // go[i,R,1]+trans[i,j,1]
          const float extra = DC[i * 8 + 4] + TR[(i * NN + j) * 2 + 0]
                              + CLt[(i + 1) * STR + j];                        // t==0 (NOCHILD) term
          IR[i * STR + j] = logadd(base, extra);
        }
      } else {
        // IL: lse_k CR[i][k] + KL[k+1][j], k = i..j-2  (valence 1 part)
        float Dv = band_lse_tile(CRt, KL, i0, s, w, w - 1, 0, 1, lane, hasD);
        const int i = i0 + d;
        if (hasD && i < s) {
          const int j = i + w;
          const float base  = Dv + DC[j * 8 + 2] + TR[(j * NN + i) * 2 + 1];   // go[j,L,1]+trans[j,i,1]
          const float extra = CRt[i * STR + (j - 1)]
                              + DC[j * 8 + 0] + TR[(j * NN + i) * 2 + 0];      // t==w-1 (NOCHILD) term
          IL[i * STR + j] = logadd(base, extra);
        }
      }
    }
    __syncthreads();

    // stage 2: complete spans KR/KL and CR/CL (consume current-width IR/IL)
    for (int job = wave; job < njobs; job += 8) {
      const int table = (job >= ntiles) ? 1 : 0;
      const int i0 = (table ? (job - ntiles) : job) << 4;
      bool hasD;
      const int d = lane & 15;
      if (table == 0) {
        // KR: lse_m IR[i][m] + CR[m][j], m = i+1..j
        float Dv = band_lse_tile(IR, CRt, i0, s, w, w, 1, 1, lane, hasD);
        const int i = i0 + d;
        if (hasD && i < s) {
          const int j = i + w;
          KR[i * STR + j]  = Dv;
          CRt[i * STR + j] = Dv + DC[i * 8 + 7];  // + stop[i,RIGHT,HASCHILD]
        }
      } else {
        // KL: lse_m CL[i][m] + IL[m][j], m = i..j-1
        float Dv = band_lse_tile(CLt, IL, i0, s, w, w, 0, 0, lane, hasD);
        const int i = i0 + d;
        if (hasD && i < s) {
          const int j = i + w;
          KL[i * STR + j]  = Dv;
          CLt[i * STR + j] = Dv + DC[j * 8 + 3];  // + stop[j,LEFT,HASCHILD]
        }
      }
    }
    __syncthreads();
  }

  if (tid == 0) {
    const int len = lens[b];
    out[b] = CRt[0 * STR + len];   // CR[0][len]
  }
}

extern "C" void kernel_launch(void* const* d_in, const int* in_sizes, int n_in,
                              void* d_out, int out_size, void* d_ws, size_t ws_size,
                              hipStream_t stream) {
  const float* trans = (const float*)d_in[0];
  const float* dec   = (const float*)d_in[1];
  const int*   lens  = (const int*)d_in[2];
  float*       out   = (float*)d_out;
  const int B = in_sizes[2];          // one sentence per workgroup
  dmv_inside_kernel<<<B, 256, 0, stream>>>(trans, dec, lens, out);
}